// CgpHmmCell_60748017434799
// MI455X (gfx1250) — compile-verified
//
#include <hip/hip_runtime.h>
#include <stdint.h>

// HMM forward pass as an associative scan of 3x3 matrix products.
//   loglik(b) = log( sum( [1,0,0] * M_0 * M_1 * ... * M_{T-1} ) ),
//   M_t = A * diag(E_t),  E_t[s] = Bmat[s,:] . x[b,t,:],  M_0 rank-1 init.
// Pass 1: 65536 threads (2048 wave32), each reduces a 128-step chunk to a
//         normalized 3x3 product + log-scale. Input streamed through LDS via
//         TRIPLE-buffered global_load_async_to_lds_b128 (ASYNCcnt), prefetch
//         distance = 2 tiles; no barriers (each lane reads only LDS bytes its
//         own wave's async ops wrote).
// Pass 2: 4096 threads fold 16 chunk records into the final loglik.

constexpr int kB     = 4096;
constexpr int kT     = 2048;
constexpr int kA     = 4;
constexpr int CHUNK  = 128;
constexpr int NC     = kT / CHUNK;    // 16 chunks per sequence
constexpr int TILE   = 8;             // timesteps staged per LDS buffer
constexpr int NBUF   = 3;             // triple buffering
constexpr int NROUND = CHUNK / TILE;  // 16 rounds per chunk
constexpr int NT     = 256;           // threads per block (8 wave32)
constexpr int WS_F   = 12;            // floats per (b,chunk) record (3x float4)

__global__ __launch_bounds__(NT) void hmm_pass1(
    const float* __restrict__ x,   // (B, T, 4) one-hot
    const float* __restrict__ tk,  // (3,)
    const float* __restrict__ ek,  // (3, 4)
    float* __restrict__ ws)        // (B * NC * WS_F)
{
  // staging: [buf][i][tid] * 16B  => 3*8*256*16 = 96 KB
  __shared__ __align__(16) float smem[NBUF * TILE * NT * 4];

  const int tid = threadIdx.x;
  const int g   = blockIdx.x * NT + tid;
  const int b   = g / NC;
  const int c   = g % NC;

  // ---- transition matrix A (6 nonzeros) ----
  const float t0 = tk[0], t1 = tk[1], t2 = tk[2];
  const float a00 = 1.f / (1.f + expf(-(1.f - t0 - t0)));
  const float a11 = 1.f / (1.f + expf(-(1.f - t1 - t1)));
  const float a21 = 1.f / (1.f + expf(-(t2 - (1.f - t2))));
  const float a01 = 1.f - a00, a12 = 1.f - a11, a22 = 1.f - a21;

  // ---- emission softmax rows ----
  float Bm[3][4];
#pragma unroll
  for (int s = 0; s < 3; ++s) {
    float e0 = ek[s * 4 + 0], e1 = ek[s * 4 + 1];
    float e2 = ek[s * 4 + 2], e3 = ek[s * 4 + 3];
    float m  = fmaxf(fmaxf(e0, e1), fmaxf(e2, e3));
    float q0 = expf(e0 - m), q1 = expf(e1 - m), q2 = expf(e2 - m), q3 = expf(e3 - m);
    float inv = 1.f / (q0 + q1 + q2 + q3);
    Bm[s][0] = q0 * inv; Bm[s][1] = q1 * inv; Bm[s][2] = q2 * inv; Bm[s][3] = q3 * inv;
  }

  const uint64_t gbase =
      (uint64_t)(uintptr_t)(x + ((size_t)b * kT + (size_t)c * CHUNK) * kA);
  const uint32_t lbase = (uint32_t)(uintptr_t)(&smem[0]) + (uint32_t)tid * 16u;

  // Issue TILE async 16B copies (one wave instruction moves 512B) for tile n.
  auto fill = [&](int n) {
    const int buf = n % NBUF;
    // Buffer being refilled was last read in the previous round; ensure our own
    // ds_load_b128s have completed before async writes can land on it.
    asm volatile("s_wait_dscnt 0" ::: "memory");
#pragma unroll
    for (int i = 0; i < TILE; ++i) {
      uint32_t laddr = lbase + (uint32_t)((buf * TILE + i) * (NT * 16));
      uint64_t gaddr = gbase + (uint64_t)((n * TILE + i) * 16);
      asm volatile("global_load_async_to_lds_b128 %0, %1, off"
                   :: "v"(laddr), "v"(gaddr) : "memory");
    }
  };

  float p00 = 1.f, p01 = 0.f, p02 = 0.f;
  float p10 = 0.f, p11 = 1.f, p12 = 0.f;
  float p20 = 0.f, p21 = 0.f, p22 = 1.f;
  float logacc = 0.f;

  fill(0);
  fill(1);
  for (int n = 0; n < NROUND; ++n) {
    // Async loads complete in order: with k tiles still outstanding after this
    // point, ASYNCcnt <= 8*k proves tile n has fully landed in LDS.
    if (n + 2 < NROUND) {
      fill(n + 2);
      asm volatile("s_wait_asynccnt 16" ::: "memory");
    } else if (n + 1 < NROUND) {
      asm volatile("s_wait_asynccnt 8" ::: "memory");
    } else {
      asm volatile("s_wait_asynccnt 0" ::: "memory");
    }
    const int buf = n % NBUF;
    const float4* sp =
        reinterpret_cast<const float4*>(smem) + (buf * TILE) * NT + tid;
#pragma unroll
    for (int i = 0; i < TILE; ++i) {
      const float4 xv = sp[i * NT];
      const float E0 = fmaf(Bm[0][3], xv.w, fmaf(Bm[0][2], xv.z, fmaf(Bm[0][1], xv.y, Bm[0][0] * xv.x)));
      const float E1 = fmaf(Bm[1][3], xv.w, fmaf(Bm[1][2], xv.z, fmaf(Bm[1][1], xv.y, Bm[1][0] * xv.x)));
      const float E2 = fmaf(Bm[2][3], xv.w, fmaf(Bm[2][2], xv.z, fmaf(Bm[2][1], xv.y, Bm[2][0] * xv.x)));
      if ((i == 0) && (n == 0) && (c == 0)) {
        // t == 0: P <- M_0 (rank-1 init absorbing alpha0 = [E0, 0, 0])
        p00 = E0;  p01 = 0.f; p02 = 0.f;
        p10 = 0.f; p11 = 0.f; p12 = 0.f;
        p20 = 0.f; p21 = 0.f; p22 = 0.f;
      } else {
        const float m00 = a00 * E0;
        const float q00 = p00 * m00;
        const float q10 = p10 * m00;
        const float q20 = p20 * m00;
        const float q01 = fmaf(p02, a21, fmaf(p01, a11, p00 * a01)) * E1;
        const float q11 = fmaf(p12, a21, fmaf(p11, a11, p10 * a01)) * E1;
        const float q21 = fmaf(p22, a21, fmaf(p21, a11, p20 * a01)) * E1;
        const float q02 = fmaf(p02, a22, p01 * a12) * E2;
        const float q12 = fmaf(p12, a22, p11 * a12) * E2;
        const float q22 = fmaf(p22, a22, p21 * a12) * E2;
        p00 = q00; p01 = q01; p02 = q02;
        p10 = q10; p11 = q11; p12 = q12;
        p20 = q20; p21 = q21; p22 = q22;
      }
    }
    // rescale every TILE steps: keeps fp32 in range, one log per 8 steps
    const float s = ((p00 + p01) + (p02 + p10)) +
                    ((p11 + p12) + (p20 + p21)) + p22;
    logacc += logf(s);
    const float inv = 1.f / s;
    p00 *= inv; p01 *= inv; p02 *= inv;
    p10 *= inv; p11 *= inv; p12 *= inv;
    p20 *= inv; p21 *= inv; p22 *= inv;
  }

  float4* w4 = reinterpret_cast<float4*>(ws + (size_t)(b * NC + c) * WS_F);
  w4[0] = make_float4(p00, p01, p02, logacc);
  w4[1] = make_float4(p10, p11, p12, 0.f);
  w4[2] = make_float4(p20, p21, p22, 0.f);
}

__global__ __launch_bounds__(256) void hmm_pass2(
    const float* __restrict__ ws, float* __restrict__ out)
{
  const int b = blockIdx.x * 256 + threadIdx.x;
  if (b >= kB) return;
  const float4* w4 = reinterpret_cast<const float4*>(ws) + (size_t)b * NC * 3;
  float v0 = 1.f, v1 = 0.f, v2 = 0.f, L = 0.f;
#pragma unroll 4
  for (int c = 0; c < NC; ++c) {
    const float4 r0 = w4[c * 3 + 0];
    const float4 r1 = w4[c * 3 + 1];
    const float4 r2 = w4[c * 3 + 2];
    L += r0.w;
    const float n0 = fmaf(v2, r2.x, fmaf(v1, r1.x, v0 * r0.x));
    const float n1 = fmaf(v2, r2.y, fmaf(v1, r1.y, v0 * r0.y));
    const float n2 = fmaf(v2, r2.z, fmaf(v1, r1.z, v0 * r0.z));
    v0 = n0; v1 = n1; v2 = n2;
  }
  out[b] = L + logf(v0 + v1 + v2);
}

extern "C" void kernel_launch(void* const* d_in, const int* in_sizes, int n_in,
                              void* d_out, int out_size, void* d_ws, size_t ws_size,
                              hipStream_t stream) {
  const float* x  = (const float*)d_in[0];  // (4096, 2048, 4) f32 one-hot
  const float* tk = (const float*)d_in[1];  // (3,)
  const float* ek = (const float*)d_in[2];  // (3, 4)
  float* out = (float*)d_out;               // (4096, 1) f32
  float* ws  = (float*)d_ws;                // needs 4096*16*12*4 = 3 MB

  hmm_pass1<<<(kB * NC) / NT, NT, 0, stream>>>(x, tk, ek, ws);
  hmm_pass2<<<kB / 256, 256, 0, stream>>>(ws, out);
}